// GRU_53386443489424
// MI455X (gfx1250) — compile-verified
//
#include <hip/hip_runtime.h>

typedef __attribute__((ext_vector_type(2))) float v2f;
typedef __attribute__((ext_vector_type(4))) float v4f;
typedef __attribute__((ext_vector_type(8))) float v8f;

// gfx1250 hardware v_tanh_f32 (TRANS32 pipe); fallback: v_exp + v_rcp, never IEEE div.
__device__ __forceinline__ float hw_tanh(float x) {
#if __has_builtin(__builtin_amdgcn_tanhf)
    return __builtin_amdgcn_tanhf(x);
#elif __has_builtin(__builtin_amdgcn_tanh_f32)
    return __builtin_amdgcn_tanh_f32(x);
#else
    float e = __expf(2.0f * x);
    return 1.0f - 2.0f * __builtin_amdgcn_rcpf(e + 1.0f);
#endif
}
__device__ __forceinline__ float fast_sigmoid(float x) {
    return 0.5f * hw_tanh(0.5f * x) + 0.5f;   // mul + tanh + fma
}

// GRU cell, one wave = one 16-row tile.
// WMMA roles: A (16x4, M x K) = gate-unit weights (LOOP-INVARIANT registers),
//             B (4x16, K x N) = data transposed, N = batch row within tile.
// C layout:   lanes 0-15, reg r  -> gate unit m=r   (z0..3 | r0..3), col = row = lane
//             lanes 16-31, reg r -> gate unit m=8+r (n0..3 | pad),   col = row = lane-16
// => lane 0-15 owns a complete output row; epilogue: 10 half-wave shuffles,
//    12 v_tanh_f32, one coalesced b128 store per row. Incremental per-lane
//    pointers (constant grid stride) avoid per-tile address rebuilds.
// Note: K-columns 10,11 of the x-GEMM are zeroed on the A (weight) side, so
// the B-side data for those columns may be arbitrary finite values -> the
// hi half's dummy in-bounds load needs NO masking.
__global__ __launch_bounds__(256) void gru_wmma_kernel(
    const float* __restrict__ x, const float* __restrict__ hidden,
    const float* __restrict__ Wzh, const float* __restrict__ bzh,
    const float* __restrict__ Wzx, const float* __restrict__ bzx,
    const float* __restrict__ Wrh, const float* __restrict__ brh,
    const float* __restrict__ Wrx, const float* __restrict__ brx,
    const float* __restrict__ Wnx, const float* __restrict__ bnx,
    const float* __restrict__ Wnh, const float* __restrict__ bnh,
    float* __restrict__ out, int B)
{
    const int lane = threadIdx.x & 31;
    const int n    = lane & 15;   // A-row (gate unit) for weights; B/C column (row) for data
    const int kh   = lane >> 4;   // K-half: 0 -> K={0,1}, 1 -> K={2,3}
    const int j    = n & 3;
    const bool mv  = (n < 12);    // gate units 12..15 are dead padding

    // ---- Loop-invariant A matrices (weights), 2 VGPRs each ----
    const float* Wh = (n < 4) ? Wzh : (n < 8) ? Wrh : Wnh;
    const float* Wx = (n < 4) ? Wzx : (n < 8) ? Wrx : Wnx;

    v2f aWh;
    aWh.x = mv ? Wh[j * 4 + 2 * kh]     : 0.0f;
    aWh.y = mv ? Wh[j * 4 + 2 * kh + 1] : 0.0f;

    v2f aWx[3];
#pragma unroll
    for (int t = 0; t < 3; ++t) {
        int k0 = 4 * t + 2 * kh;
        int k1 = k0 + 1;
        aWx[t].x = (mv && k0 < 10) ? Wx[j * 10 + k0] : 0.0f;   // zero for K >= 10
        aWx[t].y = (mv && k1 < 10) ? Wx[j * 10 + k1] : 0.0f;
    }

    // ---- Loop-invariant C-init (biases per gate unit m = r + 8*kh) ----
    v8f cInitH, cInitX;
#pragma unroll
    for (int r = 0; r < 8; ++r) {
        cInitH[r] = kh ? (r < 4 ? bnh[r] : 0.0f) : (r < 4 ? bzh[r] : brh[r - 4]);
        cInitX[r] = kh ? (r < 4 ? bnx[r] : 0.0f) : (r < 4 ? bzx[r] : brx[r - 4]);
    }

    const int Tfull  = B >> 4;                 // full 16-row tiles (no clamp needed)
    const int waveId = blockIdx.x * (blockDim.x >> 5) + (threadIdx.x >> 5);
    const int nWaves = gridDim.x * (blockDim.x >> 5);

    // ---- Incremental per-lane pointers (grid stride = nWaves*16 rows) ----
    const size_t r0 = (size_t)waveId * 16 + n;
    const float* pH  = hidden + r0 * 4  + 2 * kh;
    const float* pX  = x      + r0 * 10 + 2 * kh;
    // 3rd x-tile: lo half reads floats 8..9 of the row; hi half does an
    // in-bounds dummy read (floats 2..3) whose WMMA contribution is zero
    // because the matching A-columns (K=10,11) are zero.
    const float* pX2 = x      + r0 * 10 + (kh ? 2 : 8);
    float*       pO  = out    + r0 * 4;
    const ptrdiff_t sX = (ptrdiff_t)nWaves * 160;  // 16 rows * 10 floats
    const ptrdiff_t sH = (ptrdiff_t)nWaves * 64;   // 16 rows * 4 floats

    for (int tile = waveId; tile < Tfull; tile += nWaves) {
        v2f bHd = *(const v2f*)pH;
        v2f bX0 = *(const v2f*)pX;
        v2f bX1 = *(const v2f*)(pX + 4);
        v2f bX2 = *(const v2f*)pX2;            // no masking needed (see note)

        v8f accH = cInitH;
        v8f accX = cInitX;
        accH = __builtin_amdgcn_wmma_f32_16x16x4_f32(false, aWh,    false, bHd, (short)0, accH, false, false);
        accX = __builtin_amdgcn_wmma_f32_16x16x4_f32(false, aWx[0], false, bX0, (short)0, accX, false, false);
        accX = __builtin_amdgcn_wmma_f32_16x16x4_f32(false, aWx[1], false, bX1, (short)0, accX, false, false);
        accX = __builtin_amdgcn_wmma_f32_16x16x4_f32(false, aWx[2], false, bX2, (short)0, accX, false, false);

        // z/r gates need only the SUM of both paths (lo-lane regs 0..7).
        float s[8];
#pragma unroll
        for (int r = 0; r < 8; ++r) s[r] = accH[r] + accX[r];

        // Bring the n-path (hi half, regs 0..3) and hidden[row][2..3] to lo lanes.
        float Hn[4], Xn[4];
#pragma unroll
        for (int q = 0; q < 4; ++q) {
            Hn[q] = __shfl_xor(accH[q], 16, 32);
            Xn[q] = __shfl_xor(accX[q], 16, 32);
        }
        const float h2 = __shfl_xor(bHd.x, 16, 32);
        const float h3 = __shfl_xor(bHd.y, 16, 32);
        const float hv[4] = { bHd.x, bHd.y, h2, h3 };

        v4f o;
#pragma unroll
        for (int q = 0; q < 4; ++q) {
            const float z  = fast_sigmoid(s[q]);
            const float rr = fast_sigmoid(s[4 + q]);
            const float nn = hw_tanh(rr * Hn[q] + Xn[q]);
            o[q] = nn + z * (hv[q] - nn);     // (1-z)*n + z*h
        }

        if (lane < 16) {                      // loop-invariant store predicate
            *(v4f*)pO = o;                    // coalesced 16B/row store
        }

        pH += sH; pX += sX; pX2 += sX; pO += sH;
    }

    // ---- Scalar tail for B % 16 rows (dead when B is a multiple of 16).
    //      Loops kept rolled to minimize static code size. ----
    const int tailBase = Tfull << 4;
    if (waveId == 0 && tailBase < B) {
        const int row = tailBase + lane;
        if (row < B) {
#pragma unroll 1
            for (int u = 0; u < 4; ++u) {
                float az  = bzh[u] + bzx[u];
                float ar  = brh[u] + brx[u];
                float ahn = bnh[u];
                float axn = bnx[u];
#pragma unroll 1
                for (int k = 0; k < 4; ++k) {
                    const float hk = hidden[(size_t)row * 4 + k];
                    az  += hk * Wzh[u * 4 + k];
                    ar  += hk * Wrh[u * 4 + k];
                    ahn += hk * Wnh[u * 4 + k];
                }
#pragma unroll 1
                for (int k = 0; k < 10; ++k) {
                    const float xk = x[(size_t)row * 10 + k];
                    az  += xk * Wzx[u * 10 + k];
                    ar  += xk * Wrx[u * 10 + k];
                    axn += xk * Wnx[u * 10 + k];
                }
                const float z  = fast_sigmoid(az);
                const float rr = fast_sigmoid(ar);
                const float nn = hw_tanh(rr * ahn + axn);
                out[(size_t)row * 4 + u] = nn + z * (hidden[(size_t)row * 4 + u] - nn);
            }
        }
    }
}

extern "C" void kernel_launch(void* const* d_in, const int* in_sizes, int n_in,
                              void* d_out, int out_size, void* d_ws, size_t ws_size,
                              hipStream_t stream) {
    const float* x      = (const float*)d_in[0];
    const float* hidden = (const float*)d_in[1];
    const float* Wzh    = (const float*)d_in[2];
    const float* bzh    = (const float*)d_in[3];
    const float* Wzx    = (const float*)d_in[4];
    const float* bzx    = (const float*)d_in[5];
    const float* Wrh    = (const float*)d_in[6];
    const float* brh    = (const float*)d_in[7];
    const float* Wrx    = (const float*)d_in[8];
    const float* brx    = (const float*)d_in[9];
    const float* Wnx    = (const float*)d_in[10];
    const float* bnx    = (const float*)d_in[11];
    const float* Wnh    = (const float*)d_in[12];
    const float* bnh    = (const float*)d_in[13];
    float* out = (float*)d_out;

    const int B = in_sizes[0] / 10;   // x is [B, 10]

    // 2048 blocks x 8 wave32 waves = 16384 waves; ~16 x 16-row tiles per wave,
    // amortizing the per-wave weight/bias register setup while keeping
    // plenty of waves for HBM latency hiding.
    const int blocks  = 2048;
    const int threads = 256;
    gru_wmma_kernel<<<blocks, threads, 0, stream>>>(
        x, hidden, Wzh, bzh, Wzx, bzx, Wrh, brh, Wrx, brx, Wnx, bnx, Wnh, bnh,
        out, B);
}